// ResBlock_A_24790551232569
// MI455X (gfx1250) — compile-verified
//
#include <hip/hip_runtime.h>
#include <math.h>

// Problem constants (from reference setup_inputs)
#define NPTS   65536
#define MQ     16384
#define KNBR   32
#define PKP    15
#define HDIM   128
#define INDIM  128
#define OUTDIM 512
#define EPSBN  1e-5f
#define NSLOPE 0.1f

typedef float v2f __attribute__((ext_vector_type(2)));
typedef float v8f __attribute__((ext_vector_type(8)));

// D = A(16x4 f32) * B(4x16 f32) + C(16x16 f32), exact fp32 accumulate
__device__ __forceinline__ v8f wmma4(v2f a, v2f b, v8f c) {
    return __builtin_amdgcn_wmma_f32_16x16x4_f32(
        /*neg_a=*/false, a, /*neg_b=*/false, b,
        /*c_mod=*/(short)0, c, /*reuse_a=*/false, /*reuse_b=*/false);
}

__device__ __forceinline__ float lrelu(float v) {
    return v > 0.f ? v : NSLOPE * v;
}

// ---------------------------------------------------------------------------
__global__ void k_zero(float* __restrict__ p, int n) {
    int i = blockIdx.x * blockDim.x + threadIdx.x;
    if (i < n) p[i] = 0.f;
}

// ---------------------------------------------------------------------------
// Column sum / sum-of-squares partials (BN over rows). block (32,8).
__global__ void k_stats_partial(const float* __restrict__ x, unsigned rows, unsigned cols,
                                float* __restrict__ sum, float* __restrict__ sumsq) {
    __shared__ float ls[8][33], lq[8][33];
    unsigned tx = threadIdx.x, ty = threadIdx.y;
    unsigned c = blockIdx.x * 32u + tx;
    float s = 0.f, s2 = 0.f;
    for (unsigned r = blockIdx.y * 8u + ty; r < rows; r += 8u * gridDim.y) {
        float v = x[r * cols + c];
        s += v; s2 += v * v;
    }
    ls[ty][tx] = s; lq[ty][tx] = s2;
    __syncthreads();
    if (ty == 0) {
        for (int j = 1; j < 8; ++j) { s += ls[j][tx]; s2 += lq[j][tx]; }
        atomicAdd(&sum[c], s);
        atomicAdd(&sumsq[c], s2);
    }
}

// Fold BN into per-column affine: scale = g*rsqrt(var+eps), shift = b - mean*scale
__global__ void k_stats_final(const float* __restrict__ sum, const float* __restrict__ sumsq,
                              const float* __restrict__ g, const float* __restrict__ b,
                              float* __restrict__ scale, float* __restrict__ shift,
                              int rows, int cols) {
    int c = blockIdx.x * blockDim.x + threadIdx.x;
    if (c >= cols) return;
    float inv = 1.f / (float)rows;
    float mean = sum[c] * inv;
    float var = sumsq[c] * inv - mean * mean;
    float sc = g[c] * rsqrtf(var + EPSBN);
    scale[c] = sc;
    shift[c] = b[c] - mean * sc;
}

// ---------------------------------------------------------------------------
// C[M,N] = A[M,128] @ B[128,N]; WG = 32 rows x 128 cols, 8 waves; each wave
// computes two 16x16 tiles sharing B fragments (2 wmma per B load).
template <unsigned N>
__global__ __launch_bounds__(256) void k_gemm(const float* __restrict__ A,
                                              const float* __restrict__ B,
                                              float* __restrict__ C) {
    __shared__ float As[32][132];
    const unsigned m0 = blockIdx.x * 32u;
    const unsigned n0 = blockIdx.y * 128u;
    const unsigned tid = threadIdx.x;
    const unsigned lane = tid & 31u, wave = tid >> 5;
    for (unsigned i = tid; i < 32u * 128u; i += 256u) {
        unsigned r = i >> 7, c = i & 127u;
        As[r][c] = A[((m0 + r) << 7) + c];
    }
    __syncthreads();
    const unsigned lp = lane & 15u;
    const unsigned kh = (lane >> 4) << 1;          // 0 or 2
    const unsigned col = n0 + wave * 16u + lp;
    v8f acc0 = {}, acc1 = {};
#pragma unroll
    for (unsigned k0 = 0; k0 < 128u; k0 += 4u) {
        v2f a0, a1, b;
        b.x = B[(k0 + kh) * N + col];
        b.y = B[(k0 + kh + 1u) * N + col];
        a0.x = As[lp][k0 + kh];
        a0.y = As[lp][k0 + kh + 1u];
        a1.x = As[16u + lp][k0 + kh];
        a1.y = As[16u + lp][k0 + kh + 1u];
        acc0 = wmma4(a0, b, acc0);
        acc1 = wmma4(a1, b, acc1);
    }
#pragma unroll
    for (int j = 0; j < 8; ++j) {
        unsigned r = m0 + (unsigned)j + ((lane >> 4) << 3);
        C[r * N + col] = acc0[j];
        C[(r + 16u) * N + col] = acc1[j];
    }
}

// ---------------------------------------------------------------------------
// Fused KPConv: per WG = 16 query points.
//   infl^T (16p x 32k, p padded) -> LDS
//   Phase A: agg[p,h] = sum_k infl[k,p]*lrelu(bn1(y1[idx_k,h]))  (WMMA, gathered B)
//   Phase B: kp_out[m,:] = agg[m,:1920] @ kpw[1920,128]          (WMMA)
//   Phase C: pooled[m,h] = max_k features[idx_k,h]
// Dynamic LDS: inds 2048B + inflA 16*16*33*4 + agg 16*1921*4  (~155 KB)
__global__ __launch_bounds__(256) void k_kpconv(
        const float* __restrict__ y1, const float* __restrict__ sc1f,
        const float* __restrict__ sh1f, const float* __restrict__ pts,
        const float* __restrict__ qpts, const int* __restrict__ ninds,
        const float* __restrict__ kpts, const float* __restrict__ kpw,
        const float* __restrict__ features, float* __restrict__ kpo,
        float* __restrict__ pooled) {
    extern __shared__ char smem[];
    int*   inds   = (int*)smem;                          // 16*32
    float* inflA  = (float*)(smem + 2048);               // [16m][16p][33]
    float* aggLds = (float*)(smem + 2048 + 16*16*33*4);  // [16m][1921]

    const unsigned m0 = blockIdx.x * 16u;
    const unsigned tid = threadIdx.x;
    const unsigned lane = tid & 31u, wave = tid >> 5;

    // load neighbor indices (rows are contiguous: base = m0*32)
    for (unsigned i = tid; i < 16u * KNBR; i += 256u)
        inds[i] = ninds[(m0 << 5) + i];

    // influence weights, stored transposed [m][p][k], row p=15 zero-padded
    for (unsigned i = tid; i < 16u * KNBR; i += 256u) {
        unsigned mm = i >> 5, k = i & 31u;
        unsigned idx = (unsigned)inds[i];
        float px = pts[idx * 3u + 0u] - qpts[(m0 + mm) * 3u + 0u];
        float py = pts[idx * 3u + 1u] - qpts[(m0 + mm) * 3u + 1u];
        float pz = pts[idx * 3u + 2u] - qpts[(m0 + mm) * 3u + 2u];
#pragma unroll
        for (unsigned p = 0; p < 16u; ++p) {
            float w = 0.f;
            if (p < PKP) {
                float dx = px - kpts[p * 3u + 0u];
                float dy = py - kpts[p * 3u + 1u];
                float dz = pz - kpts[p * 3u + 2u];
                float d = sqrtf(dx * dx + dy * dy + dz * dz);
                w = fmaxf(0.f, 1.f - d);   // SIGMA = 1.0
            }
            inflA[mm * (16u * 33u) + p * 33u + k] = w;
        }
    }
    __syncthreads();

    const unsigned lp = lane & 15u;
    const unsigned kh = (lane >> 4) << 1;

    // ---- Phase A: per-point (16p x 32k) @ (32k x 128h), B gathered from y1 ----
    for (unsigned mm = wave; mm < 16u; mm += 8u) {
        const int* mi = inds + mm * KNBR;
#pragma unroll
        for (unsigned ht = 0; ht < 8u; ++ht) {
            unsigned col = ht * 16u + lp;
            float s1 = sc1f[col], h1 = sh1f[col];
            v8f acc = {};
#pragma unroll
            for (unsigned ks = 0; ks < 8u; ++ks) {
                unsigned k0 = ks * 4u;
                v2f a, b;
                a.x = inflA[mm * (16u * 33u) + lp * 33u + k0 + kh];
                a.y = inflA[mm * (16u * 33u) + lp * 33u + k0 + kh + 1u];
                unsigned i0 = (unsigned)mi[k0 + kh];
                unsigned i1 = (unsigned)mi[k0 + kh + 1u];
                b.x = lrelu(y1[(i0 << 7) + col] * s1 + h1);
                b.y = lrelu(y1[(i1 << 7) + col] * s1 + h1);
                acc = wmma4(a, b, acc);
            }
#pragma unroll
            for (int j = 0; j < 8; ++j) {
                unsigned p = (unsigned)j + ((lane >> 4) << 3);
                if (p < PKP) aggLds[mm * 1921u + (p << 7) + col] = acc[j];
            }
        }
    }
    __syncthreads();

    // ---- Phase B: [16m x 1920] @ kpw[1920 x 128]; wave w -> out cols w*16.. ----
    {
        const unsigned colh = wave * 16u + lp;
        v8f acc = {};
#pragma unroll 8
        for (unsigned kk = 0; kk < PKP * HDIM; kk += 4u) {
            v2f a, b;
            a.x = aggLds[lp * 1921u + kk + kh];
            a.y = aggLds[lp * 1921u + kk + kh + 1u];
            b.x = kpw[((kk + kh) << 7) + colh];
            b.y = kpw[((kk + kh + 1u) << 7) + colh];
            acc = wmma4(a, b, acc);
        }
#pragma unroll
        for (int j = 0; j < 8; ++j) {
            unsigned r = m0 + (unsigned)j + ((lane >> 4) << 3);
            kpo[(r << 7) + colh] = acc[j];
        }
    }

    // ---- Phase C: shortcut max-pool over neighbors ----
    for (unsigned i = tid; i < 16u * INDIM; i += 256u) {
        unsigned mm = i >> 7, h = i & 127u;
        const int* mi = inds + mm * KNBR;
        float mx = -3.402823466e38f;
#pragma unroll 4
        for (unsigned k = 0; k < KNBR; ++k)
            mx = fmaxf(mx, features[((unsigned)mi[k] << 7) + h]);
        pooled[((m0 + mm) << 7) + h] = mx;
    }
}

// ---------------------------------------------------------------------------
// out = lrelu( lrelu(bn2(kp_out)) @ w3 + b3 + bn_sc(sc_pre) )
// WG = 32 rows x 128 cols; each wave: two 16x16 tiles sharing B fragments.
__global__ __launch_bounds__(256) void k_final(
        const float* __restrict__ kpo, const float* __restrict__ sc2,
        const float* __restrict__ sh2, const float* __restrict__ w3,
        const float* __restrict__ b3, const float* __restrict__ scpre,
        const float* __restrict__ scsc, const float* __restrict__ shsc,
        float* __restrict__ out) {
    __shared__ float As[32][132];
    const unsigned m0 = blockIdx.x * 32u;
    const unsigned n0 = blockIdx.y * 128u;
    const unsigned tid = threadIdx.x;
    const unsigned lane = tid & 31u, wave = tid >> 5;
    for (unsigned i = tid; i < 32u * 128u; i += 256u) {
        unsigned r = i >> 7, c = i & 127u;
        float v = kpo[((m0 + r) << 7) + c] * sc2[c] + sh2[c];
        As[r][c] = lrelu(v);
    }
    __syncthreads();
    const unsigned lp = lane & 15u;
    const unsigned kh = (lane >> 4) << 1;
    const unsigned col = n0 + wave * 16u + lp;
    v8f acc0 = {}, acc1 = {};
#pragma unroll
    for (unsigned k0 = 0; k0 < 128u; k0 += 4u) {
        v2f a0, a1, b;
        b.x = w3[((k0 + kh) << 9) + col];
        b.y = w3[((k0 + kh + 1u) << 9) + col];
        a0.x = As[lp][k0 + kh];
        a0.y = As[lp][k0 + kh + 1u];
        a1.x = As[16u + lp][k0 + kh];
        a1.y = As[16u + lp][k0 + kh + 1u];
        acc0 = wmma4(a0, b, acc0);
        acc1 = wmma4(a1, b, acc1);
    }
    float bb = b3[col], ss = scsc[col], hh = shsc[col];
#pragma unroll
    for (int j = 0; j < 8; ++j) {
        unsigned r = m0 + (unsigned)j + ((lane >> 4) << 3);
        float v0 = acc0[j] + bb + scpre[(r << 9) + col] * ss + hh;
        out[(r << 9) + col] = lrelu(v0);
        unsigned r1 = r + 16u;
        float v1 = acc1[j] + bb + scpre[(r1 << 9) + col] * ss + hh;
        out[(r1 << 9) + col] = lrelu(v1);
    }
}

// ---------------------------------------------------------------------------
extern "C" void kernel_launch(void* const* d_in, const int* in_sizes, int n_in,
                              void* d_out, int out_size, void* d_ws, size_t ws_size,
                              hipStream_t stream) {
    (void)in_sizes; (void)n_in; (void)out_size; (void)ws_size;
    const float* features = (const float*)d_in[0];
    const float* points   = (const float*)d_in[1];
    const float* qpts     = (const float*)d_in[2];
    const int*   ninds    = (const int*)  d_in[3];
    const float* kpts     = (const float*)d_in[4];
    const float* w1       = (const float*)d_in[5];
    const float* g1       = (const float*)d_in[6];
    const float* b1       = (const float*)d_in[7];
    const float* kpw      = (const float*)d_in[8];
    const float* g2       = (const float*)d_in[9];
    const float* b2       = (const float*)d_in[10];
    const float* w3       = (const float*)d_in[11];
    const float* b3       = (const float*)d_in[12];
    const float* wsc      = (const float*)d_in[13];
    const float* gsc      = (const float*)d_in[14];
    const float* bsc      = (const float*)d_in[15];
    float* out = (float*)d_out;

    // workspace layout (floats)
    float* ws     = (float*)d_ws;
    float* y1     = ws;                                   // 65536*128
    float* kpo    = y1    + (size_t)NPTS * HDIM;          // 16384*128
    float* pooled = kpo   + (size_t)MQ * HDIM;            // 16384*128
    float* scpre  = pooled + (size_t)MQ * INDIM;          // 16384*512
    float* stats  = scpre + (size_t)MQ * OUTDIM;
    float* sum1 = stats;        float* sq1 = sum1 + 128;
    float* sum2 = sq1 + 128;    float* sq2 = sum2 + 128;
    float* sumsc = sq2 + 128;   float* sqsc = sumsc + 512;
    float* sc1 = sqsc + 512;    float* sh1 = sc1 + 128;
    float* sc2 = sh1 + 128;     float* sh2 = sc2 + 128;
    float* scsc = sh2 + 128;    float* shsc = scsc + 512;

    // 0) zero stat accumulators (sum1..sqsc = 1536 floats)
    k_zero<<<6, 256, 0, stream>>>(stats, 1536);

    // 1) y1 = features @ w1
    k_gemm<HDIM><<<dim3(NPTS / 32, 1), 256, 0, stream>>>(features, w1, y1);

    // 2) BN1 stats -> affine coefficients
    k_stats_partial<<<dim3(HDIM / 32, 64), dim3(32, 8), 0, stream>>>(y1, NPTS, HDIM, sum1, sq1);
    k_stats_final<<<1, 128, 0, stream>>>(sum1, sq1, g1, b1, sc1, sh1, NPTS, HDIM);

    // 3) fused KPConv (+ shortcut max-pool)
    size_t lds = 2048 + (size_t)16 * 16 * 33 * 4 + (size_t)16 * 1921 * 4;  // ~155 KB
    k_kpconv<<<MQ / 16, 256, lds, stream>>>(y1, sc1, sh1, points, qpts, ninds,
                                            kpts, kpw, features, kpo, pooled);

    // 4) BN2 stats
    k_stats_partial<<<dim3(HDIM / 32, 64), dim3(32, 8), 0, stream>>>(kpo, MQ, HDIM, sum2, sq2);
    k_stats_final<<<1, 128, 0, stream>>>(sum2, sq2, g2, b2, sc2, sh2, MQ, HDIM);

    // 5) shortcut GEMM: sc_pre = pooled @ wsc
    k_gemm<OUTDIM><<<dim3(MQ / 32, OUTDIM / 128), 256, 0, stream>>>(pooled, wsc, scpre);

    // 6) BN_sc stats
    k_stats_partial<<<dim3(OUTDIM / 32, 64), dim3(32, 8), 0, stream>>>(scpre, MQ, OUTDIM, sumsc, sqsc);
    k_stats_final<<<OUTDIM / 128, 128, 0, stream>>>(sumsc, sqsc, gsc, bsc, scsc, shsc, MQ, OUTDIM);

    // 7) final GEMM + residual + lrelu
    k_final<<<dim3(MQ / 32, OUTDIM / 128), 256, 0, stream>>>(kpo, sc2, sh2, w3, b3,
                                                             scpre, scsc, shsc, out);
}